// _HighOrderFactorizationMachineModel_33182917329443
// MI455X (gfx1250) — compile-verified
//
#include <hip/hip_runtime.h>
#include <stdint.h>

// High-order FM (order 3), B=16384, F=10, D=64, row=128 floats.
// Strategy: async-stage the 260 hot embedding rows (133KB) into LDS,
// per-wave 16-sample tiles, power-sum accumulation in registers,
// lane-reduction via V_WMMA_F32_16X16X4_F32 (T[16x32] x ones).

#define FIELDS   10
#define ROW      128     // EMBED_DIM * (ORDER-1)
#define HOT      26      // indices are in [0,26); min field dim is 26 -> always in bounds
#define TILE     16      // samples per wave (WMMA M dimension)
#define WAVES    4
#define TPB      (WAVES * 32)
#define TSTRIDE  34      // even -> 8B-aligned f32x2 rows, odd-ish bank spread

typedef float f32x2 __attribute__((ext_vector_type(2)));
typedef float f32x4 __attribute__((ext_vector_type(4)));
typedef float f32x8 __attribute__((ext_vector_type(8)));

__global__ __launch_bounds__(TPB) void hofm3_kernel(
    const int* __restrict__ x,        // [B, FIELDS]
    const int* __restrict__ offsets,  // [FIELDS]
    const float* __restrict__ W_emb,  // [total, ROW]
    const float* __restrict__ W_lin,  // [total]
    const float* __restrict__ bias,   // [1]
    float* __restrict__ out,          // [B]
    int batch)
{
  __shared__ __align__(16) float cache[FIELDS * HOT * ROW];   // 133,120 B hot rows
  __shared__ __align__(16) float T[WAVES][TILE][TSTRIDE];     // per-wave partial matrix
  __shared__ float lin_cache[FIELDS * HOT];                   // hot W_lin entries

  const int tid  = threadIdx.x;
  const int lane = tid & 31;
  const int wid  = tid >> 5;

  // ---- Stage hot embedding rows into LDS via CDNA5 async global->LDS copy ----
  {
    const uint32_t cbase = (uint32_t)(uintptr_t)&cache[0];
    const unsigned long long gbase = (unsigned long long)(uintptr_t)W_emb;
    for (int it = 0; it < (FIELDS * HOT * ROW / 4) / TPB; ++it) {  // 65 iters
      int e   = it * TPB + tid;              // float4 index into cache
      int f   = e / (HOT * ROW / 4);         // field (div by 832)
      int rem = e - f * (HOT * ROW / 4);
      int r   = rem >> 5;                    // hot row within field
      int d4  = rem & 31;                    // float4 within row
      unsigned long long g = gbase
          + ((unsigned long long)(uint32_t)(offsets[f] + r) << 9)  // row*512B
          + ((uint32_t)d4 << 4);
      uint32_t l = cbase + ((uint32_t)e << 4);
      asm volatile("global_load_async_to_lds_b128 %0, %1, off"
                   :: "v"(l), "v"(g) : "memory");
    }
    for (int t = tid; t < FIELDS * HOT; t += TPB) {
      int f = t / HOT;
      lin_cache[t] = W_lin[offsets[f] + (t - f * HOT)];
    }
    asm volatile("s_wait_asynccnt 0x0" ::: "memory");
  }
  __syncthreads();

  const int tile_id = blockIdx.x * WAVES + wid;
  const int s0 = tile_id * TILE;
  if (s0 + TILE <= batch) {
    const int   off_l = (lane < FIELDS) ? offsets[lane] : 0;
    const float bias0 = bias[0];

    for (int s = 0; s < TILE; ++s) {
      const int sample = s0 + s;
      int xi = 0;
      if (lane < FIELDS) xi = x[sample * FIELDS + lane];
      float lin = 0.f;
      if (lane < FIELDS)
        lin = (xi < HOT) ? lin_cache[lane * HOT + xi] : W_lin[xi + off_l];

      // lane l owns dims 4l..4l+3; lanes 0-15 = FM half, 16-31 = ANOVA half
      float p1[4] = {0.f, 0.f, 0.f, 0.f};
      float p2[4] = {0.f, 0.f, 0.f, 0.f};
      float p3[4] = {0.f, 0.f, 0.f, 0.f};
      #pragma unroll
      for (int f = 0; f < FIELDS; ++f) {
        const int xv = __shfl(xi, f, 32);    // v_readlane -> scalar row id
        f32x4 ev;
        if (xv < HOT) {                      // wave-uniform branch, always hot here
          ev = *(const f32x4*)&cache[(f * HOT + xv) * ROW + lane * 4];
        } else {                             // correctness fallback: global gather
          const int row = xv + __shfl(off_l, f, 32);
          ev = *(const f32x4*)(W_emb + ((unsigned long long)(uint32_t)row << 7)
                               + lane * 4);
        }
        #pragma unroll
        for (int j = 0; j < 4; ++j) {
          const float ex = ev[j];
          const float e2 = ex * ex;
          p1[j] += ex;
          p2[j] += e2;
          p3[j] = fmaf(ex, e2, p3[j]);
        }
      }
      // Newton's identities: e2 = (p1^2-p2)/2 ; e3 = (p1^3 - 3 p1 p2 + 2 p3)/6
      float fm = 0.f, an = 0.f;
      #pragma unroll
      for (int j = 0; j < 4; ++j) {
        fm += 0.5f * (p1[j] * p1[j] - p2[j]);
        an += (p1[j] * (p1[j] * p1[j] - 3.f * p2[j]) + 2.f * p3[j]) * (1.f / 6.f);
      }
      float P = (lane < 16) ? fm : an;
      P += lin;
      if (lane == 0) P += bias0;
      T[wid][s][lane] = P;                   // row s of the 16x32 partial matrix
    }
    asm volatile("" ::: "memory");           // same-wave LDS ops are HW in-order

    // ---- y[m] = sum_k T[m][k] as 8 accumulating WMMA f32 16x16x4, B = ones ----
    f32x8 c = {0.f, 0.f, 0.f, 0.f, 0.f, 0.f, 0.f, 0.f};
    const f32x2 bones = {1.f, 1.f};
    const int m     = lane & 15;
    const int khalf = (lane >> 4) << 1;      // lanes 0-15 -> K {0,1}; 16-31 -> K {2,3}
    #pragma unroll
    for (int i = 0; i < 8; ++i) {
      const int k = i * 4 + khalf;
      f32x2 a = *(const f32x2*)&T[wid][m][k];
      c = __builtin_amdgcn_wmma_f32_16x16x4_f32(false, a, false, bones,
                                                (short)0, c, false, false);
    }
    #pragma unroll
    for (int j = 0; j < 8; ++j) c[j] = fmaxf(c[j], 0.f);
    // C layout: lane 0 VGPR r -> y[r], lane 16 VGPR r -> y[8+r]
    if ((lane & 15) == 0) {
      float* o = out + s0 + ((lane >> 4) << 3);
      f32x4 lo = {c[0], c[1], c[2], c[3]};
      f32x4 hi = {c[4], c[5], c[6], c[7]};
      *(f32x4*)o = lo;
      *(f32x4*)(o + 4) = hi;
    }
  }
}

extern "C" void kernel_launch(void* const* d_in, const int* in_sizes, int n_in,
                              void* d_out, int out_size, void* d_ws, size_t ws_size,
                              hipStream_t stream) {
  const int*   x       = (const int*)d_in[0];
  const int*   offsets = (const int*)d_in[1];
  const float* W_emb   = (const float*)d_in[2];
  const float* W_lin   = (const float*)d_in[3];
  const float* bias    = (const float*)d_in[4];
  float* out = (float*)d_out;

  const int batch  = in_sizes[0] / FIELDS;            // 16384
  const int tiles  = (batch + TILE - 1) / TILE;       // 1024
  const int blocks = (tiles + WAVES - 1) / WAVES;     // 256

  hofm3_kernel<<<blocks, TPB, 0, stream>>>(x, offsets, W_emb, W_lin, bias, out, batch);
}